// GAT54_32306744000781
// MI455X (gfx1250) — compile-verified
//
#include <hip/hip_runtime.h>
#include <hip/hip_bf16.h>
#include <math.h>

// ---------------------------------------------------------------------------
// 2-layer GAT (heads=1) for MI455X / gfx1250.
//   Layer: h = X @ W  (WMMA f32 16x16x4, K templated & fully unrolled)
//   -> edge softmax (3 passes, u32-keyed atomic max + f32 atomic adds, all
//      dst-indexed arrays L2-resident) -> bias (+ELU between layers).
// ---------------------------------------------------------------------------

typedef __attribute__((ext_vector_type(2))) float v2f;
typedef __attribute__((ext_vector_type(8))) float v8f;

#define F_IN 54
#define FH   16           // hidden == out channels
#define NEG_SLOPE 0.2f
#define ENC_NEG_INF 0x007FFFFFu   // order-preserving encoding of -inf

// order-preserving float <-> uint mapping (segment-max == atomic u32 max)
__device__ __forceinline__ unsigned fenc(float f) {
  unsigned u = __float_as_uint(f);
  return (u & 0x80000000u) ? ~u : (u | 0x80000000u);
}
__device__ __forceinline__ float fdec(unsigned k) {
  return (k & 0x80000000u) ? __uint_as_float(k & 0x7FFFFFFFu)
                           : __uint_as_float(~k);
}

// --------------------------- dense GEMM via WMMA ---------------------------
// Out[N x 16] = X[N x K] @ W[K x 16], one wave per 16-row tile.
// A (16x4 f32): lanes 0-15 hold K={k,k+1}, lanes 16-31 K={k+2,k+3}, M=lane&15
// B (4x16 f32): VGPR0 = row k(+2), VGPR1 = row k+1(+3), N=lane&15
// C/D: VGPR r -> M=r (lanes 0-15) / M=r+8 (lanes 16-31), N=lane&15
template <int K>
__global__ void gat_gemm_wmma(const float* __restrict__ X,
                              const float* __restrict__ W,
                              float* __restrict__ Out, int N) {
  const int wid  = (blockIdx.x * blockDim.x + threadIdx.x) >> 5;
  const int lane = threadIdx.x & 31;
  const int row0 = wid << 4;
  if (row0 >= N) return;               // wave-uniform branch
  const int m    = lane & 15;
  const int hi   = lane >> 4;          // 0 | 1
  const int koff = hi * 2;
  int row = row0 + m;
  if (row >= N) row = N - 1;           // clamp (loads stay valid)
  const float* __restrict__ Xr = X + (size_t)row * K;

  v8f c = {0.f, 0.f, 0.f, 0.f, 0.f, 0.f, 0.f, 0.f};

  constexpr int KFULL = (K / 4) * 4;   // iterations with all lanes in-bounds
#pragma unroll
  for (int k = 0; k + 4 <= K; k += 4) {
    const int k0 = k + koff;
    const v2f a = *(const v2f*)(Xr + k0);          // 8B-aligned pair load
    v2f b;
    b.x = W[k0 * FH + m];
    b.y = W[(k0 + 1) * FH + m];
    c = __builtin_amdgcn_wmma_f32_16x16x4_f32(false, a, false, b,
                                              (short)0, c, false, false);
  }
  if constexpr (K % 4 != 0) {
    // tail: unconditional loads from clamped addresses + value selects
    // (keeps EXEC all-1s, no saveexec branches)
    const int k0  = KFULL + koff;
    const bool v0 = (k0 < K), v1 = (k0 + 1 < K);
    const int i0  = v0 ? k0 : 0;
    const int i1  = v1 ? (k0 + 1) : 0;
    const float xa0 = Xr[i0], xa1 = Xr[i1];
    const float wb0 = W[i0 * FH + m], wb1 = W[i1 * FH + m];
    v2f a, b;
    a.x = v0 ? xa0 : 0.f;  a.y = v1 ? xa1 : 0.f;
    b.x = v0 ? wb0 : 0.f;  b.y = v1 ? wb1 : 0.f;
    c = __builtin_amdgcn_wmma_f32_16x16x4_f32(false, a, false, b,
                                              (short)0, c, false, false);
  }

  const int rb = hi * 8;
  float* __restrict__ Ot = Out + (size_t)(row0 + rb) * FH + m;
  if (row0 + 16 <= N) {                // full tile: branchless stores
#pragma unroll
    for (int r = 0; r < 8; ++r) Ot[r * FH] = c[r];
  } else {                             // ragged last tile
#pragma unroll
    for (int r = 0; r < 8; ++r)
      if (row0 + rb + r < N) Ot[r * FH] = c[r];
  }
}

// ------------------- per-node attention prep + zero agg --------------------
__global__ void node_prep(const float* __restrict__ H,
                          const float* __restrict__ a_s,
                          const float* __restrict__ a_d,
                          float* __restrict__ as, float* __restrict__ ad,
                          unsigned* __restrict__ mx, float* __restrict__ den,
                          float* __restrict__ agg, int N) {
  const int i = blockIdx.x * blockDim.x + threadIdx.x;
  if (i >= N) return;
  float s = 0.f, d = 0.f;
#pragma unroll
  for (int f = 0; f < FH; ++f) {
    const float hv = H[(size_t)i * FH + f];
    s += hv * a_s[f];
    d += hv * a_d[f];
    agg[(size_t)i * FH + f] = 0.f;
  }
  as[i]  = s;
  ad[i]  = d;
  mx[i]  = ENC_NEG_INF;
  den[i] = 0.f;
}

// ---------------- pass 1: logits + segment max (u32 atomic) ----------------
__global__ void edge_max(const int* __restrict__ ei,
                         const float* __restrict__ as,
                         const float* __restrict__ ad,
                         float* __restrict__ ebuf, unsigned* __restrict__ mx,
                         int E, int N) {
  const int id = blockIdx.x * blockDim.x + threadIdx.x;
  if (id >= E + N) return;
  const int src = (id < E) ? ei[id]     : (id - E);   // self-loops appended
  const int dst = (id < E) ? ei[E + id] : (id - E);
  float e = as[src] + ad[dst];
  e = (e > 0.f) ? e : NEG_SLOPE * e;                  // LeakyReLU
  ebuf[id] = e;
  atomicMax(&mx[dst], fenc(e));
}

// ---------------- pass 2: exp(e - max) + segment sum -----------------------
__global__ void edge_expsum(const int* __restrict__ ei,
                            float* __restrict__ ebuf,
                            const unsigned* __restrict__ mx,
                            float* __restrict__ den, int E, int N) {
  const int id = blockIdx.x * blockDim.x + threadIdx.x;
  if (id >= E + N) return;
  const int dst = (id < E) ? ei[E + id] : (id - E);
  const float t = __expf(ebuf[id] - fdec(mx[dst]));
  ebuf[id] = t;
  atomicAdd(&den[dst], t);
}

// -------- pass 3: alpha = e/den, agg[dst] += alpha * h[src] (16 thr/edge) --
__global__ void edge_aggr(const int* __restrict__ ei,
                          const float* __restrict__ ebuf,
                          const float* __restrict__ den,
                          const float* __restrict__ H,
                          float* __restrict__ agg, int E, int N) {
  const long long gid = (long long)blockIdx.x * blockDim.x + threadIdx.x;
  const long long tot = (long long)(E + N) * FH;
  if (gid >= tot) return;
  const int id  = (int)(gid >> 4);
  const int f   = (int)(gid & 15);
  const int src = (id < E) ? ei[id]     : (id - E);
  const int dst = (id < E) ? ei[E + id] : (id - E);
  const float alpha = ebuf[id] / den[dst];
  atomicAdd(&agg[(size_t)dst * FH + f], alpha * H[(size_t)src * FH + f]);
}

// ----------------------- bias (+ optional ELU) -----------------------------
__global__ void bias_act(const float* __restrict__ agg,
                         const float* __restrict__ b,
                         float* __restrict__ out, int NF, int elu) {
  const int i = blockIdx.x * blockDim.x + threadIdx.x;
  if (i >= NF) return;
  float v = agg[i] + b[i & (FH - 1)];
  if (elu) v = (v > 0.f) ? v : (__expf(v) - 1.f);
  out[i] = v;
}

// ---------------------------------------------------------------------------
extern "C" void kernel_launch(void* const* d_in, const int* in_sizes, int n_in,
                              void* d_out, int out_size, void* d_ws,
                              size_t ws_size, hipStream_t stream) {
  const float* x   = (const float*)d_in[0];
  const int*   ei  = (const int*)d_in[1];     // edge_index, flat [2,E]
  const float* W1  = (const float*)d_in[2];
  const float* a1s = (const float*)d_in[3];
  const float* a1d = (const float*)d_in[4];
  const float* b1  = (const float*)d_in[5];
  const float* W2  = (const float*)d_in[6];
  const float* a2s = (const float*)d_in[7];
  const float* a2d = (const float*)d_in[8];
  const float* b2  = (const float*)d_in[9];
  float* out = (float*)d_out;

  const int N  = in_sizes[0] / F_IN;
  const int E  = in_sizes[1] / 2;
  const int Et = E + N;

  // workspace carve (~34 MB total)
  char* p = (char*)d_ws;
  float*    h1  = (float*)p;    p += (size_t)N * FH * sizeof(float);
  float*    h2  = (float*)p;    p += (size_t)N * FH * sizeof(float);
  float*    agg = (float*)p;    p += (size_t)N * FH * sizeof(float);
  float*    as  = (float*)p;    p += (size_t)N * sizeof(float);
  float*    ad  = (float*)p;    p += (size_t)N * sizeof(float);
  float*    den = (float*)p;    p += (size_t)N * sizeof(float);
  unsigned* mx  = (unsigned*)p; p += (size_t)N * sizeof(unsigned);
  float*    eb  = (float*)p;    p += (size_t)Et * sizeof(float);
  (void)ws_size; (void)n_in; (void)out_size;

  const int B = 256;
  auto G = [](long long t, int b) { return dim3((unsigned)((t + b - 1) / b)); };
  const long long gemmThreads = (long long)((N + 15) / 16) * 32;

  // ----- layer 1 -----
  gat_gemm_wmma<F_IN><<<G(gemmThreads, B), B, 0, stream>>>(x, W1, h1, N);
  node_prep    <<<G(N, B),  B, 0, stream>>>(h1, a1s, a1d, as, ad, mx, den, agg, N);
  edge_max     <<<G(Et, B), B, 0, stream>>>(ei, as, ad, eb, mx, E, N);
  edge_expsum  <<<G(Et, B), B, 0, stream>>>(ei, eb, mx, den, E, N);
  edge_aggr    <<<G((long long)Et * FH, B), B, 0, stream>>>(ei, eb, den, h1, agg, E, N);
  bias_act     <<<G((long long)N * FH, B), B, 0, stream>>>(agg, b1, h1, N * FH, 1);

  // ----- layer 2 -----
  gat_gemm_wmma<FH><<<G(gemmThreads, B), B, 0, stream>>>(h1, W2, h2, N);
  node_prep    <<<G(N, B),  B, 0, stream>>>(h2, a2s, a2d, as, ad, mx, den, agg, N);
  edge_max     <<<G(Et, B), B, 0, stream>>>(ei, as, ad, eb, mx, E, N);
  edge_expsum  <<<G(Et, B), B, 0, stream>>>(ei, eb, mx, den, E, N);
  edge_aggr    <<<G((long long)Et * FH, B), B, 0, stream>>>(ei, eb, den, h2, agg, E, N);
  bias_act     <<<G((long long)N * FH, B), B, 0, stream>>>(agg, b2, out, N * FH, 0);
}